// AgriMatcher_46188078301772
// MI455X (gfx1250) — compile-verified
//
#include <hip/hip_runtime.h>
#include <math.h>

// Problem constants (match reference)
#define BB 2
#define NN 4096
#define CC 128
#define HH 64
#define WW 64
#define TK 8
#define RAD 3
#define SINK_ITERS 5
#define INV_TEMP 20.0f

typedef __attribute__((ext_vector_type(2))) float v2f;
typedef __attribute__((ext_vector_type(8))) float v8f;

// ---------- online log-sum-exp helpers ----------
__device__ __forceinline__ void lse_accum(float& mx, float& s, float z) {
  if (z > mx) { s = s * expf(mx - z) + 1.0f; mx = z; }
  else        { s += expf(z - mx); }
}
__device__ __forceinline__ void lse_combine(float& m1, float& s1, float m2, float s2) {
  if (m2 > m1)               { s1 = s2 + s1 * expf(m1 - m2); m1 = m2; }
  else if (m1 > -INFINITY)   { s1 += s2 * expf(m2 - m1); }
}

// ---------- 1) inverse row norms (wave32 shuffle reduction) ----------
__global__ void knorm(const float* __restrict__ fa, const float* __restrict__ fb,
                      float* __restrict__ invn) {
  int wid  = (blockIdx.x * blockDim.x + threadIdx.x) >> 5;  // one wave per row
  int lane = threadIdx.x & 31;
  if (wid >= 2 * BB * NN) return;
  const float* f = (wid < BB * NN) ? fa : fb;
  int row = (wid < BB * NN) ? wid : wid - BB * NN;
  const float4* p = (const float4*)(f + (size_t)row * CC);
  float4 x = p[lane];                       // 32 lanes x 4 = 128 elems
  float ss = x.x * x.x + x.y * x.y + x.z * x.z + x.w * x.w;
  for (int off = 16; off > 0; off >>= 1) ss += __shfl_xor(ss, off, 32);
  if (lane == 0) invn[wid] = 1.0f / fmaxf(sqrtf(ss), 1e-12f);
}

// ---------- 2) zero-fill ----------
__global__ void kzero(float* __restrict__ p, int n) {
  int i = blockIdx.x * blockDim.x + threadIdx.x;
  if (i < n) p[i] = 0.0f;
}

// ---------- 3) FP32 WMMA GEMM: raw_sim = normA @ normB^T ----------
// 64x32 macro-tile per wave = 4x2 grid of 16x16 WMMA tiles.
// Per K-step (K=4): 6 x global_load_b64 + 6 x v_pk_mul_f32 -> 8 x
// V_WMMA_F32_16X16X4_F32. 8 accs (64 VGPRs) + 6 fragments stays under the
// 128-VGPR budget -> no scratch spills (4x4 blocking spilled last round).
// Per-lane 32-bit A/B layout (ISA 7.12.2): row/col = lane%16, K = 2*(lane/16)+j.
__global__ void kgemm(const float* __restrict__ fa, const float* __restrict__ fb,
                      const float* __restrict__ invn, float* __restrict__ sim) {
  int lane = threadIdx.x & 31;
  int wave = threadIdx.x >> 5;
  int b  = blockIdx.z;
  int m0 = blockIdx.y * 64;
  int n0 = (blockIdx.x * 4 + wave) * 32;
  int r  = lane & 15;
  int kb = (lane >> 4) << 1;  // 0 or 2

  const float* Abase = fa + (size_t)(b * NN + m0 + r) * CC + kb;
  const float* Bbase = fb + (size_t)(b * NN + n0 + r) * CC + kb;
  float ia[4], ib[2];
#pragma unroll
  for (int i = 0; i < 4; ++i) ia[i] = invn[b * NN + m0 + i * 16 + r];
#pragma unroll
  for (int j = 0; j < 2; ++j) ib[j] = invn[BB * NN + b * NN + n0 + j * 16 + r];

  v8f acc[4][2] = {};
#pragma unroll 2
  for (int c = 0; c < CC; c += 4) {
    v2f av[4], bv[2];
#pragma unroll
    for (int i = 0; i < 4; ++i) {
      float2 a2 = *(const float2*)(Abase + (size_t)i * 16 * CC + c);
      av[i].x = a2.x * ia[i]; av[i].y = a2.y * ia[i];
    }
#pragma unroll
    for (int j = 0; j < 2; ++j) {
      float2 b2 = *(const float2*)(Bbase + (size_t)j * 16 * CC + c);
      bv[j].x = b2.x * ib[j]; bv[j].y = b2.y * ib[j];
    }
#pragma unroll
    for (int i = 0; i < 4; ++i)
#pragma unroll
      for (int j = 0; j < 2; ++j)
        acc[i][j] = __builtin_amdgcn_wmma_f32_16x16x4_f32(false, av[i], false, bv[j],
                                                          (short)0, acc[i][j], false, false);
  }
  // D layout: VGPR v -> M = v + 8*(lane/16), N = lane%16
  int mb = (lane >> 4) << 3;
#pragma unroll
  for (int i = 0; i < 4; ++i)
#pragma unroll
    for (int j = 0; j < 2; ++j) {
      float* orow = sim + (size_t)b * NN * NN +
                    (size_t)(m0 + i * 16 + mb) * NN + n0 + j * 16 + r;
#pragma unroll
      for (int vv = 0; vv < 8; ++vv) orow[(size_t)vv * NN] = acc[i][j][vv];
    }
}

// ---------- 4) Sinkhorn u-update (rows) ----------
__global__ void ku(const float* __restrict__ sim, const float* __restrict__ v,
                   float* __restrict__ u) {
  int m = blockIdx.x, b = blockIdx.y, t = threadIdx.x;
  const float* vb = v + b * (NN + 1);
  float mx = -INFINITY, s = 0.0f;
  if (m < NN) {
    const float* row = sim + (size_t)b * NN * NN + (size_t)m * NN;
    for (int n = t; n < NN; n += 256) lse_accum(mx, s, row[n] * INV_TEMP + vb[n]);
    if (t == 0) lse_accum(mx, s, vb[NN]);            // dustbin column, Z=0
  } else {                                           // dustbin row, Z=0
    for (int n = t; n <= NN; n += 256) lse_accum(mx, s, vb[n]);
  }
  __shared__ float sm[256], ss[256];
  sm[t] = mx; ss[t] = s; __syncthreads();
  for (int off = 128; off > 0; off >>= 1) {
    if (t < off) {
      float m1 = sm[t], s1 = ss[t];
      lse_combine(m1, s1, sm[t + off], ss[t + off]);
      sm[t] = m1; ss[t] = s1;
    }
    __syncthreads();
  }
  if (t == 0) {
    float lse = sm[0] + logf(ss[0]);
    float logmu = (m < NN) ? -logf(2.0f * NN) : (logf((float)NN) - logf(2.0f * NN));
    u[b * (NN + 1) + m] = logmu - lse;
  }
}

// ---------- 5) Sinkhorn v-update (columns, 32-col x 8-row tiles) ----------
__global__ void kv(const float* __restrict__ sim, const float* __restrict__ u,
                   float* __restrict__ v) {
  int b = blockIdx.y, t = threadIdx.x;
  int c = t & 31, rg = t >> 5;
  int n = blockIdx.x * 32 + c;
  const float* ub = u + b * (NN + 1);
  float mx = -INFINITY, s = 0.0f;
  if (n <= NN) {
    if (n < NN) {
      const float* col = sim + (size_t)b * NN * NN + n;
      for (int m = rg; m < NN; m += 8)
        lse_accum(mx, s, col[(size_t)m * NN] * INV_TEMP + ub[m]);
    } else {                                         // dustbin column, Z=0
      for (int m = rg; m < NN; m += 8) lse_accum(mx, s, ub[m]);
    }
    if (rg == 0) lse_accum(mx, s, ub[NN]);           // dustbin row, Z=0
  }
  __shared__ float sm[256], ss[256];
  sm[t] = mx; ss[t] = s; __syncthreads();
  for (int off = 4; off > 0; off >>= 1) {
    if (rg < off) {
      float m1 = sm[t], s1 = ss[t];
      lse_combine(m1, s1, sm[t + off * 32], ss[t + off * 32]);
      sm[t] = m1; ss[t] = s1;
    }
    __syncthreads();
  }
  if (rg == 0 && n <= NN) {
    float lse = sm[t] + logf(ss[t]);
    float lognu = (n < NN) ? -logf(2.0f * NN) : (logf((float)NN) - logf(2.0f * NN));
    v[b * (NN + 1) + n] = lognu - lse;
  }
}

// ---------- 6) per-row: rowsum, entropy, top-8 ----------
__global__ void kfin(const float* __restrict__ sim, const float* __restrict__ u,
                     const float* __restrict__ v, float* __restrict__ ent_out,
                     float* __restrict__ tv_out, int* __restrict__ ti_out) {
  int m = blockIdx.x, b = blockIdx.y, t = threadIdx.x;
  const float* row = sim + (size_t)b * NN * NN + (size_t)m * NN;
  float um = u[b * (NN + 1) + m];
  const float* vb = v + b * (NN + 1);

  float lv[TK]; int li[TK];
#pragma unroll
  for (int k = 0; k < TK; ++k) { lv[k] = -INFINITY; li[k] = 0x7FFFFFFF; }
  float sum = 0.0f;
  for (int n = t; n < NN; n += 256) {
    float p = expf(row[n] * INV_TEMP + um + vb[n]);
    sum += p;
    if (p > lv[TK - 1]) {
      int k = TK - 1;
      while (k > 0 && lv[k - 1] < p) { lv[k] = lv[k - 1]; li[k] = li[k - 1]; --k; }
      lv[k] = p; li[k] = n;
    }
  }
  __shared__ float sred[256];
  __shared__ float rowsum_sh;
  sred[t] = sum; __syncthreads();
  for (int off = 128; off > 0; off >>= 1) {
    if (t < off) sred[t] += sred[t + off];
    __syncthreads();
  }
  if (t == 0) rowsum_sh = sred[0];
  __syncthreads();
  float rs = rowsum_sh + 1e-8f;

  float e = 0.0f;
  for (int n = t; n < NN; n += 256) {
    float p = expf(row[n] * INV_TEMP + um + vb[n]);
    float q = p / rs;
    e -= q * logf(q + 1e-8f);
  }
  sred[t] = e; __syncthreads();
  for (int off = 128; off > 0; off >>= 1) {
    if (t < off) sred[t] += sred[t + off];
    __syncthreads();
  }
  if (t == 0) ent_out[b * NN + m] = sred[0];

  // block-wide top-8 merge: 8 rounds of argmax (tie -> smaller column index)
  __shared__ float sv[256]; __shared__ int si[256]; __shared__ int st[256];
  __shared__ int winner;
  int ptr = 0;
  for (int k = 0; k < TK; ++k) {
    sv[t] = (ptr < TK) ? lv[ptr] : -INFINITY;
    si[t] = (ptr < TK) ? li[ptr] : 0x7FFFFFFF;
    st[t] = t;
    __syncthreads();
    for (int off = 128; off > 0; off >>= 1) {
      if (t < off) {
        float v2 = sv[t + off]; int i2 = si[t + off];
        if (v2 > sv[t] || (v2 == sv[t] && i2 < si[t])) {
          sv[t] = v2; si[t] = i2; st[t] = st[t + off];
        }
      }
      __syncthreads();
    }
    if (t == 0) {
      tv_out[((size_t)b * NN + m) * TK + k] = sv[0];
      ti_out[((size_t)b * NN + m) * TK + k] = (si[0] == 0x7FFFFFFF) ? 0 : si[0];
      winner = st[0];
    }
    __syncthreads();
    if (t == winner) ++ptr;
    __syncthreads();
  }
}

// ---------- 7) gather candidate positions + displacement planes ----------
__global__ void kgather(const int* __restrict__ ti, const float* __restrict__ posA,
                        const float* __restrict__ posB, float* __restrict__ tpos,
                        float* __restrict__ disp) {
  int id = blockIdx.x * blockDim.x + threadIdx.x;
  if (id >= BB * NN * TK) return;
  int k = id % TK, n = (id / TK) % NN, b = id / (TK * NN);
  int idx = ti[id];
  float px = posB[((size_t)b * NN + idx) * 2 + 0];
  float py = posB[((size_t)b * NN + idx) * 2 + 1];
  tpos[(size_t)id * 2 + 0] = px;
  tpos[(size_t)id * 2 + 1] = py;
  float ax = posA[((size_t)b * NN + n) * 2 + 0];
  float ay = posA[((size_t)b * NN + n) * 2 + 1];
  size_t base = ((size_t)b * TK + k) * 2 * (HH * WW);   // [B][K][2][H][W]
  disp[base + n]           = px - ax;
  disp[base + HH * WW + n] = py - ay;
}

// ---------- 8) 7x7 neighborhood variance -> geometric score ----------
__global__ void kgeo(const float* __restrict__ disp, float* __restrict__ geo) {
  int id = blockIdx.x * blockDim.x + threadIdx.x;
  if (id >= BB * TK * HH * WW) return;
  int w = id % WW, h = (id / WW) % HH, k = (id / (WW * HH)) % TK, b = id / (WW * HH * TK);
  const float* p0 = disp + ((size_t)b * TK + k) * 2 * (HH * WW);
  const float* p1 = p0 + HH * WW;
  float sum = 0.0f, sq = 0.0f;
  for (int di = -RAD; di <= RAD; ++di) {
    int hh = h + di;
    if (hh < 0 || hh >= HH) continue;       // zero padding contributes 0
    for (int dj = -RAD; dj <= RAD; ++dj) {
      int ww = w + dj;
      if (ww < 0 || ww >= WW) continue;
      float a = p0[hh * WW + ww], c2 = p1[hh * WW + ww];
      sum += a + c2; sq += a * a + c2 * c2;
    }
  }
  float var = (sq - sum * sum / 98.0f) / 97.0f;   // 2*49 elems, ddof=1
  float score = 1.0f / (1.0f + var * 100.0f);
  int n = h * WW + w;
  geo[((size_t)b * NN + n) * TK + k] = score;
}

// ---------- 9) softmax refinement ----------
__global__ void krefine(const float* __restrict__ tv, const float* __restrict__ geo,
                        const float* __restrict__ tpos, float* __restrict__ warp_out) {
  int id = blockIdx.x * blockDim.x + threadIdx.x;
  if (id >= BB * NN) return;
  const float* tvp = tv + (size_t)id * TK;
  const float* gp  = geo + (size_t)id * TK;
  const float* pp  = tpos + (size_t)id * TK * 2;
  float c[TK], mx = -INFINITY;
#pragma unroll
  for (int k = 0; k < TK; ++k) { c[k] = (tvp[k] + 1.5f * gp[k]) * INV_TEMP; mx = fmaxf(mx, c[k]); }
  float sw = 0.0f, wk[TK];
#pragma unroll
  for (int k = 0; k < TK; ++k) { wk[k] = expf(c[k] - mx); sw += wk[k]; }
  float rx = 0.0f, ry = 0.0f;
#pragma unroll
  for (int k = 0; k < TK; ++k) {
    float wnorm = wk[k] / sw;
    rx += pp[k * 2 + 0] * wnorm;
    ry += pp[k * 2 + 1] * wnorm;
  }
  warp_out[(size_t)id * 2 + 0] = rx;
  warp_out[(size_t)id * 2 + 1] = ry;
}

extern "C" void kernel_launch(void* const* d_in, const int* in_sizes, int n_in,
                              void* d_out, int out_size, void* d_ws, size_t ws_size,
                              hipStream_t stream) {
  (void)in_sizes; (void)n_in; (void)out_size; (void)ws_size;
  const float* featA = (const float*)d_in[0];
  const float* featB = (const float*)d_in[1];
  const float* posA  = (const float*)d_in[2];
  const float* posB  = (const float*)d_in[3];

  float* out      = (float*)d_out;
  float* warp_out = out;                                   // [B,N,2]
  float* ent_out  = out + (size_t)BB * NN * 2;             // [B,N]
  float* sim_out  = out + (size_t)BB * NN * 2 + (size_t)BB * NN;  // [B,N,N]

  // scratch layout (~2 MB of floats)
  float* f     = (float*)d_ws;
  float* invn  = f;                            // 2*B*N
  float* u     = invn + 2 * BB * NN;           // B*(N+1)
  float* v     = u + BB * (NN + 1);            // B*(N+1)
  float* tvals = v + BB * (NN + 1);            // B*N*K
  float* tpos  = tvals + BB * NN * TK;         // B*N*K*2
  float* geo   = tpos + (size_t)BB * NN * TK * 2;  // B*N*K
  float* disp  = geo + BB * NN * TK;           // B*K*2*H*W
  int*   tidx  = (int*)(disp + (size_t)BB * TK * 2 * HH * WW);  // B*N*K

  // 1) inverse norms (one wave32 per row)
  knorm<<<dim3(2 * BB * NN / 8), 256, 0, stream>>>(featA, featB, invn);
  // 2) u = v = 0 (u and v are contiguous)
  int nzero = 2 * BB * (NN + 1);
  kzero<<<(nzero + 255) / 256, 256, 0, stream>>>(u, nzero);
  // 3) cosine-similarity GEMM via V_WMMA_F32_16X16X4_F32, 64x32 per wave
  kgemm<<<dim3(NN / 128, NN / 64, BB), 128, 0, stream>>>(featA, featB, invn, sim_out);
  // 4) 5 log-domain Sinkhorn iterations (matrix stays L2-resident: 134 MB < 192 MB)
  for (int it = 0; it < SINK_ITERS; ++it) {
    ku<<<dim3(NN + 1, BB), 256, 0, stream>>>(sim_out, v, u);
    kv<<<dim3((NN + 32) / 32, BB), 256, 0, stream>>>(sim_out, u, v);
  }
  // 5) entropy + top-8 per row
  kfin<<<dim3(NN, BB), 256, 0, stream>>>(sim_out, u, v, ent_out, tvals, tidx);
  // 6) gather positions & displacements
  kgather<<<(BB * NN * TK + 255) / 256, 256, 0, stream>>>(tidx, posA, posB, tpos, disp);
  // 7) local geometric variance score
  kgeo<<<(BB * TK * HH * WW + 255) / 256, 256, 0, stream>>>(disp, geo);
  // 8) softmax-weighted refinement
  krefine<<<(BB * NN + 255) / 256, 256, 0, stream>>>(tvals, geo, tpos, warp_out);
}